// RigidFAPE_44006234914911
// MI455X (gfx1250) — compile-verified
//
#include <hip/hip_runtime.h>
#include <math.h>

// ---------------------------------------------------------------------------
// RigidFAPE on MI455X (gfx1250, wave32)
//   pass 1: streaming covariance + centroid sums  (V_WMMA_F32_16X16X4_F32)
//   pass 2: per-batch 3x3 Kabsch SVD (Jacobi on H^T H)
//   pass 3: streaming clamped point error (L2-resident re-read, 192MB L2)
//   pass 4: deterministic final reduction -> scalar mean
// Frames inputs (268 MB) are intentionally never read.
// ---------------------------------------------------------------------------

typedef __attribute__((ext_vector_type(2))) float v2f;
typedef __attribute__((ext_vector_type(8))) float v8f;

constexpr int   BATCH = 256;
constexpr int   NPTS  = 8192;
constexpr int   SPLIT = 4;               // blocks per batch in streaming passes
constexpr int   CH    = NPTS / SPLIT;    // samples per block = 2048
constexpr int   TPB   = 256;             // 8 waves (wave32)
constexpr float CLAMPV = 10.0f;

// workspace layout in floats
constexpr int OFF_COV = 0;                         // BATCH*SPLIT * 16
constexpr int OFF_RT  = BATCH * SPLIT * 16;        // + BATCH * 12
constexpr int OFF_ERR = OFF_RT + BATCH * 12;       // + BATCH*SPLIT
// total = 20480 floats = 80 KB

// ---------------------------------------------------------------------------
// Pass 1: per (batch, split) block: 9 product sums via WMMA + 6 centroid sums
// ---------------------------------------------------------------------------
__global__ __launch_bounds__(TPB) void cov_kernel(const float* __restrict__ pred,
                                                  const float* __restrict__ tru,
                                                  float* __restrict__ ws) {
  __shared__ float pl[TPB][4];     // staged pred tile; column 3 is a zero pad
  __shared__ float ql[TPB][4];     // staged true tile; column 3 is a zero pad
  __shared__ float prod[8][9];     // per-wave 3x3 partials
  __shared__ float red[TPB];

  const int tid  = threadIdx.x;
  const int wave = tid >> 5;
  const int lane = tid & 31;
  const int h    = lane >> 4;      // half-wave: selects K pair {0,1} vs {2,3}
  const int m    = lane & 15;      // M index for A, N index for B
  const int mm   = (m < 3) ? m : 3;   // lanes m>=3 read the zero pad column

  const int blk  = blockIdx.x;
  const int b    = blk / SPLIT;
  const int sp   = blk % SPLIT;
  const int base = (b * NPTS + sp * CH) * 3;

  float sax = 0.f, say = 0.f, saz = 0.f;
  float sbx = 0.f, sby = 0.f, sbz = 0.f;
  v8f acc = {};   // 16x16 f32 accumulator; rows/cols 0..2 are the real data

  for (int t = 0; t < CH / TPB; ++t) {
    const int s = base + (t * TPB + tid) * 3;
    const float px = pred[s], py = pred[s + 1], pz = pred[s + 2];
    const float qx = tru[s],  qy = tru[s + 1],  qz = tru[s + 2];
    sax += px; say += py; saz += pz;
    sbx += qx; sby += qy; sbz += qz;
    pl[tid][0] = px; pl[tid][1] = py; pl[tid][2] = pz; pl[tid][3] = 0.f;
    ql[tid][0] = qx; ql[tid][1] = qy; ql[tid][2] = qz; ql[tid][3] = 0.f;
    __syncthreads();

    // wave w consumes samples [w*32, w*32+32) of this tile: 8 WMMAs of K=4.
    // A (16x4 f32): lanes 0-15 -> VGPR0=K0,VGPR1=K1 ; lanes 16-31 -> K2,K3 ; M=lane%16
    // B (4x16 f32): mirrored with lane%16 = N
    // Lanes m>=3 carry zeros (pad column); their tile rows/cols are never read.
    for (int c = 0; c < 8; ++c) {
      const int s0 = wave * 32 + c * 4 + 2 * h;
      v2f av, bv;
      av.x = pl[s0][mm];      av.y = pl[s0 + 1][mm];
      bv.x = ql[s0][mm];      bv.y = ql[s0 + 1][mm];
      acc = __builtin_amdgcn_wmma_f32_16x16x4_f32(
          /*neg_a=*/false, av, /*neg_b=*/false, bv,
          /*c_mod=*/(short)0, acc, /*reuse_a=*/false, /*reuse_b=*/false);
    }
    __syncthreads();
  }

  // C/D layout: VGPR r, lanes 0-15 -> D[M=r][N=lane]. 3x3 block: lanes 0..2, VGPR 0..2.
  if (lane < 3) {
    prod[wave][0 * 3 + lane] = acc[0];
    prod[wave][1 * 3 + lane] = acc[1];
    prod[wave][2 * 3 + lane] = acc[2];
  }
  __syncthreads();

  float* out = ws + OFF_COV + blk * 16;
  if (tid < 9) {
    float v = 0.f;
    for (int w = 0; w < 8; ++w) v += prod[w][tid];
    out[tid] = v;                                   // [0..8]  = sum p_i * q_j
  }

  float vals[6] = {sax, say, saz, sbx, sby, sbz};
  for (int k = 0; k < 6; ++k) {
    __syncthreads();
    red[tid] = vals[k];
    __syncthreads();
    for (int off = TPB / 2; off > 0; off >>= 1) {
      if (tid < off) red[tid] += red[tid + off];
      __syncthreads();
    }
    if (tid == 0) out[9 + k] = red[0];              // [9..11]=sumA, [12..14]=sumB
  }
}

// ---------------------------------------------------------------------------
// Pass 2: one thread per batch: Kabsch via Jacobi eigen of H^T H
// ---------------------------------------------------------------------------
__global__ __launch_bounds__(TPB) void kabsch_kernel(float* __restrict__ ws) {
  const int b = blockIdx.x * blockDim.x + threadIdx.x;
  if (b >= BATCH) return;

  float P[9]  = {0.f};
  float SA[3] = {0.f}, SB[3] = {0.f};
  for (int s = 0; s < SPLIT; ++s) {
    const float* c = ws + OFF_COV + (b * SPLIT + s) * 16;
    for (int k = 0; k < 9; ++k) P[k] += c[k];
    for (int k = 0; k < 3; ++k) { SA[k] += c[9 + k]; SB[k] += c[12 + k]; }
  }
  const float invN = 1.0f / (float)NPTS;
  float cA[3], cB[3];
  for (int k = 0; k < 3; ++k) { cA[k] = SA[k] * invN; cB[k] = SB[k] * invN; }

  float H[3][3];
  for (int i = 0; i < 3; ++i)
    for (int j = 0; j < 3; ++j)
      H[i][j] = P[i * 3 + j] - (float)NPTS * cA[i] * cB[j];

  // K = H^T H (symmetric PSD)
  float A[3][3];
  for (int i = 0; i < 3; ++i)
    for (int j = 0; j < 3; ++j) {
      float s = 0.f;
      for (int k = 0; k < 3; ++k) s += H[k][i] * H[k][j];
      A[i][j] = s;
    }

  float V[3][3] = {{1.f, 0.f, 0.f}, {0.f, 1.f, 0.f}, {0.f, 0.f, 1.f}};
  for (int sweep = 0; sweep < 12; ++sweep) {
    for (int pi = 0; pi < 3; ++pi) {
      const int p = (pi == 2) ? 1 : 0;
      const int q = (pi == 0) ? 1 : 2;
      const float apq = A[p][q];
      if (fabsf(apq) < 1e-25f) continue;
      const float tau = (A[q][q] - A[p][p]) / (2.0f * apq);
      const float tt  = (tau >= 0.f) ? 1.0f / (tau + sqrtf(1.0f + tau * tau))
                                     : 1.0f / (tau - sqrtf(1.0f + tau * tau));
      const float c = 1.0f / sqrtf(1.0f + tt * tt);
      const float s = tt * c;
      for (int k = 0; k < 3; ++k) {
        const float akp = A[k][p], akq = A[k][q];
        A[k][p] = c * akp - s * akq;  A[k][q] = s * akp + c * akq;
      }
      for (int k = 0; k < 3; ++k) {
        const float apk = A[p][k], aqk = A[q][k];
        A[p][k] = c * apk - s * aqk;  A[q][k] = s * apk + c * aqk;
      }
      for (int k = 0; k < 3; ++k) {
        const float vkp = V[k][p], vkq = V[k][q];
        V[k][p] = c * vkp - s * vkq;  V[k][q] = s * vkp + c * vkq;
      }
    }
  }

  float w[3] = {A[0][0], A[1][1], A[2][2]};
  int idx[3] = {0, 1, 2};
  if (w[idx[0]] < w[idx[1]]) { int t = idx[0]; idx[0] = idx[1]; idx[1] = t; }
  if (w[idx[0]] < w[idx[2]]) { int t = idx[0]; idx[0] = idx[2]; idx[2] = t; }
  if (w[idx[1]] < w[idx[2]]) { int t = idx[1]; idx[1] = idx[2]; idx[2] = t; }

  float Vs[3][3], U[3][3];  // columns = singular vectors, descending sigma
  for (int c2 = 0; c2 < 3; ++c2) {
    const int j = idx[c2];
    const float v0 = V[0][j], v1 = V[1][j], v2 = V[2][j];
    Vs[0][c2] = v0; Vs[1][c2] = v1; Vs[2][c2] = v2;
    const float u0 = H[0][0] * v0 + H[0][1] * v1 + H[0][2] * v2;
    const float u1 = H[1][0] * v0 + H[1][1] * v1 + H[1][2] * v2;
    const float u2 = H[2][0] * v0 + H[2][1] * v1 + H[2][2] * v2;
    const float sig = sqrtf(fmaxf(w[j], 0.f));
    const float inv = 1.0f / fmaxf(sig, 1e-20f);
    U[0][c2] = u0 * inv; U[1][c2] = u1 * inv; U[2][c2] = u2 * inv;
  }

  // R = Vs * U^T ; flip smallest-sigma column of Vs if det(R) < 0
  float R[3][3];
  for (int i = 0; i < 3; ++i)
    for (int j = 0; j < 3; ++j)
      R[i][j] = Vs[i][0] * U[j][0] + Vs[i][1] * U[j][1] + Vs[i][2] * U[j][2];
  const float det = R[0][0] * (R[1][1] * R[2][2] - R[1][2] * R[2][1])
                  - R[0][1] * (R[1][0] * R[2][2] - R[1][2] * R[2][0])
                  + R[0][2] * (R[1][0] * R[2][1] - R[1][1] * R[2][0]);
  if (det < 0.f)
    for (int i = 0; i < 3; ++i)
      for (int j = 0; j < 3; ++j)
        R[i][j] -= 2.0f * Vs[i][2] * U[j][2];

  float t3[3];
  for (int i = 0; i < 3; ++i)
    t3[i] = cB[i] - (R[i][0] * cA[0] + R[i][1] * cA[1] + R[i][2] * cA[2]);

  float* rt = ws + OFF_RT + b * 12;
  for (int i = 0; i < 3; ++i)
    for (int j = 0; j < 3; ++j) rt[i * 3 + j] = R[i][j];
  rt[9] = t3[0]; rt[10] = t3[1]; rt[11] = t3[2];
}

// ---------------------------------------------------------------------------
// Pass 3: clamped point error, per-block partial sums (L2-resident re-read)
// ---------------------------------------------------------------------------
__global__ __launch_bounds__(TPB) void err_kernel(const float* __restrict__ pred,
                                                  const float* __restrict__ tru,
                                                  float* __restrict__ ws) {
  __shared__ float red[TPB];
  __shared__ float rt[12];
  const int tid = threadIdx.x;
  const int blk = blockIdx.x;
  const int b   = blk / SPLIT;
  const int sp  = blk % SPLIT;
  if (tid < 12) rt[tid] = ws[OFF_RT + b * 12 + tid];
  __syncthreads();
  const float R00 = rt[0], R01 = rt[1], R02 = rt[2];
  const float R10 = rt[3], R11 = rt[4], R12 = rt[5];
  const float R20 = rt[6], R21 = rt[7], R22 = rt[8];
  const float t0 = rt[9], t1 = rt[10], t2 = rt[11];

  const int base = (b * NPTS + sp * CH) * 3;
  float part = 0.f;
  for (int t = 0; t < CH / TPB; ++t) {
    const int s = base + (t * TPB + tid) * 3;
    const float px = pred[s], py = pred[s + 1], pz = pred[s + 2];
    const float qx = tru[s],  qy = tru[s + 1],  qz = tru[s + 2];
    // reference applies pred @ R (row-vector convention), then +t
    const float ax = px * R00 + py * R10 + pz * R20 + t0 - qx;
    const float ay = px * R01 + py * R11 + pz * R21 + t1 - qy;
    const float az = px * R02 + py * R12 + pz * R22 + t2 - qz;
    const float d = sqrtf(ax * ax + ay * ay + az * az);
    part += fminf(d, CLAMPV);
  }
  red[tid] = part;
  __syncthreads();
  for (int off = TPB / 2; off > 0; off >>= 1) {
    if (tid < off) red[tid] += red[tid + off];
    __syncthreads();
  }
  if (tid == 0) ws[OFF_ERR + blk] = red[0];
}

// ---------------------------------------------------------------------------
// Pass 4: deterministic final reduction -> global mean
// ---------------------------------------------------------------------------
__global__ __launch_bounds__(TPB) void finish_kernel(const float* __restrict__ ws,
                                                     float* __restrict__ out) {
  __shared__ float red[TPB];
  const int tid = threadIdx.x;
  float v = 0.f;
  for (int k = tid; k < BATCH * SPLIT; k += TPB) v += ws[OFF_ERR + k];
  red[tid] = v;
  __syncthreads();
  for (int off = TPB / 2; off > 0; off >>= 1) {
    if (tid < off) red[tid] += red[tid + off];
    __syncthreads();
  }
  if (tid == 0) out[0] = red[0] / (float)(BATCH * NPTS);
}

// ---------------------------------------------------------------------------
extern "C" void kernel_launch(void* const* d_in, const int* in_sizes, int n_in,
                              void* d_out, int out_size, void* d_ws, size_t ws_size,
                              hipStream_t stream) {
  (void)in_sizes; (void)n_in; (void)out_size; (void)ws_size;
  // setup_inputs order: [0]=pred_frames (unused), [1]=true_frames (unused),
  //                     [2]=pred_pos, [3]=true_pos
  const float* pred = (const float*)d_in[2];
  const float* tru  = (const float*)d_in[3];
  float* ws  = (float*)d_ws;
  float* out = (float*)d_out;

  cov_kernel   <<<BATCH * SPLIT, TPB, 0, stream>>>(pred, tru, ws);
  kabsch_kernel<<<1,             TPB, 0, stream>>>(ws);
  err_kernel   <<<BATCH * SPLIT, TPB, 0, stream>>>(pred, tru, ws);
  finish_kernel<<<1,             TPB, 0, stream>>>(ws, out);
}